// HBS_28338194219185
// MI455X (gfx1250) — compile-verified
//
#include <hip/hip_runtime.h>
#include <hip/hip_bf16.h>

typedef __attribute__((ext_vector_type(2))) float v2f;
typedef __attribute__((ext_vector_type(8))) float v8f;

#define N_CELLS 50000
#define N_EDGES 1600000
#define C 64
#define NEG_SLOPE 0.2f

// ---------------------------------------------------------------------------
// Kernel 1: msg = X @ W  via V_WMMA_F32_16X16X4_F32 (fp32-exact matrix pipe).
// Block = 128 threads = 4 waves. Each block computes a 16-row stripe of msg;
// wave w computes the 16x16 tile at columns [16w, 16w+16). K=64 -> 16 WMMAs.
// Layouts (CDNA5 ISA):
//   A 16x4 f32 : lane L (m = L&15, h = L>>4) holds A[m][k0+2h], A[m][k0+2h+1]
//   B 4x16 f32 : lane L holds B[k0+2h][n], B[k0+2h+1][n],  n = n0 + (L&15)
//   C/D 16x16  : lane L, vgpr r holds D[r + 8h][n]
// ---------------------------------------------------------------------------
__global__ __launch_bounds__(128) void hbs_gemm_wmma(
    const float* __restrict__ X, const float* __restrict__ W,
    float* __restrict__ msg) {
  const int lane   = threadIdx.x & 31;
  const int wave   = threadIdx.x >> 5;          // 0..3 -> N tile
  const int row0   = blockIdx.x * 16;
  const int m      = lane & 15;
  const int h      = lane >> 4;                 // lane-half select
  const int khalf  = h * 2;
  const int ncol   = wave * 16 + (lane & 15);

  const float* xrow = X + (size_t)(row0 + m) * C;

  v8f acc = {};
#pragma unroll
  for (int k0 = 0; k0 < C; k0 += 4) {
    const int ka = k0 + khalf;
    v2f a;
    a.x = xrow[ka];
    a.y = xrow[ka + 1];
    v2f b;
    b.x = W[(size_t)ka * C + ncol];
    b.y = W[(size_t)(ka + 1) * C + ncol];
    acc = __builtin_amdgcn_wmma_f32_16x16x4_f32(
        /*neg_a=*/false, a, /*neg_b=*/false, b,
        /*c_mod=*/(short)0, acc, /*reuse_a=*/false, /*reuse_b=*/false);
  }

#pragma unroll
  for (int r = 0; r < 8; ++r) {
    const int row = row0 + r + 8 * h;
    msg[(size_t)row * C + ncol] = acc[r];
  }
}

// ---------------------------------------------------------------------------
// Kernel 2: per-cell attention logits  alpha_src = msg.a[:64], alpha_dst = msg.a[64:]
// One thread per cell; 256B sequential read per thread (L2-resident msg).
// ---------------------------------------------------------------------------
__global__ __launch_bounds__(256) void hbs_alpha(
    const float* __restrict__ msg, const float* __restrict__ a,
    float* __restrict__ alpha_src, float* __restrict__ alpha_dst) {
  const int c = blockIdx.x * blockDim.x + threadIdx.x;
  if (c >= N_CELLS) return;
  const float* row = msg + (size_t)c * C;
  float s0 = 0.f, s1 = 0.f;
#pragma unroll 8
  for (int i = 0; i < C; ++i) {
    const float v = row[i];
    s0 += v * a[i];
    s1 += v * a[C + i];
  }
  alpha_src[c] = s0;
  alpha_dst[c] = s1;
}

// ---------------------------------------------------------------------------
// Kernel 3: zero-fill (graph-capture safe; harness poisons buffers with 0xAA).
// ---------------------------------------------------------------------------
__global__ __launch_bounds__(256) void hbs_zero(float* __restrict__ p, int n) {
  const int i = blockIdx.x * blockDim.x + threadIdx.x;
  if (i < n) p[i] = 0.f;
}

// ---------------------------------------------------------------------------
// Kernel 4: per-edge score e = leaky_relu(alpha_src[s] + alpha_dst[d]) and
// atomic row sums (f32 atomics land in L2; row_sum is 200 KB, fully resident).
// ---------------------------------------------------------------------------
__global__ __launch_bounds__(256) void hbs_edge_score(
    const int* __restrict__ src, const int* __restrict__ dst,
    const float* __restrict__ alpha_src, const float* __restrict__ alpha_dst,
    float* __restrict__ e_arr, float* __restrict__ row_sum) {
  const int e = blockIdx.x * blockDim.x + threadIdx.x;
  if (e >= N_EDGES) return;
  const int s = src[e];
  const int d = dst[e];
  const float v  = alpha_src[s] + alpha_dst[d];
  const float el = v > 0.f ? v : NEG_SLOPE * v;
  e_arr[e] = el;
  atomicAdd(&row_sum[s], el);
}

// ---------------------------------------------------------------------------
// Kernel 5: one wave (32 lanes) per edge. scale = e/row_sum[s] * edge_value.
// Lane l gathers msg[d][2l:2l+2] as a coalesced float2 (256B per wave, random
// row but msg fits in L2), then two f32 atomic adds into out[s].
// ---------------------------------------------------------------------------
__global__ __launch_bounds__(256) void hbs_edge_accum(
    const int* __restrict__ src, const int* __restrict__ dst,
    const float* __restrict__ e_arr, const float* __restrict__ row_sum,
    const float* __restrict__ edge_values, const float* __restrict__ msg,
    float* __restrict__ out) {
  const int gtid = blockIdx.x * blockDim.x + threadIdx.x;
  const int wid  = gtid >> 5;   // edge index (wave32)
  const int lane = gtid & 31;
  if (wid >= N_EDGES) return;
  const int s = src[wid];
  const int d = dst[wid];
  const float scale = e_arr[wid] / row_sum[s] * edge_values[wid];
  const float2 mv = ((const float2*)(msg + (size_t)d * C))[lane];
  float* orow = out + (size_t)s * C + 2 * lane;
  atomicAdd(orow + 0, scale * mv.x);
  atomicAdd(orow + 1, scale * mv.y);
}

// ---------------------------------------------------------------------------
extern "C" void kernel_launch(void* const* d_in, const int* in_sizes, int n_in,
                              void* d_out, int out_size, void* d_ws, size_t ws_size,
                              hipStream_t stream) {
  const float* x_source    = (const float*)d_in[0];                 // [50000,64]
  const int*   edge_index  = (const int*)d_in[1];                   // [2,E]
  const float* edge_values = (const float*)d_in[2];                 // [E]
  const float* W           = (const float*)d_in[3];                 // [64,64]
  const float* a           = (const float*)d_in[4];                 // [128]
  float* out = (float*)d_out;                                       // [50000,64]

  const int* src = edge_index;             // row 0
  const int* dst = edge_index + N_EDGES;   // row 1

  // Workspace layout (floats): msg | alpha_src | alpha_dst | row_sum | e
  float* msg       = (float*)d_ws;
  float* alpha_src = msg + (size_t)N_CELLS * C;
  float* alpha_dst = alpha_src + N_CELLS;
  float* row_sum   = alpha_dst + N_CELLS;
  float* e_arr     = row_sum + N_CELLS;

  // 1) msg = X @ W (WMMA fp32).  50000/16 = 3125 row tiles.
  hbs_gemm_wmma<<<N_CELLS / 16, 128, 0, stream>>>(x_source, W, msg);

  // 2) alphas.
  hbs_alpha<<<(N_CELLS + 255) / 256, 256, 0, stream>>>(msg, a, alpha_src, alpha_dst);

  // 3) zero accumulators (out and row_sum) every call.
  hbs_zero<<<((int)((size_t)N_CELLS * C) + 255) / 256, 256, 0, stream>>>(out, N_CELLS * C);
  hbs_zero<<<(N_CELLS + 255) / 256, 256, 0, stream>>>(row_sum, N_CELLS);

  // 4) edge scores + row sums.
  hbs_edge_score<<<(N_EDGES + 255) / 256, 256, 0, stream>>>(
      src, dst, alpha_src, alpha_dst, e_arr, row_sum);

  // 5) normalized scatter-accumulate: one wave per edge.
  const long long threads = (long long)N_EDGES * 32;
  hbs_edge_accum<<<(int)((threads + 255) / 256), 256, 0, stream>>>(
      src, dst, e_arr, row_sum, edge_values, msg, out);
}